// DGCNNEncoder_37701222924949
// MI455X (gfx1250) — compile-verified
//
#include <hip/hip_runtime.h>

#define DEV __device__ __forceinline__

typedef float v2f __attribute__((ext_vector_type(2)));
typedef float v4f __attribute__((ext_vector_type(4)));
typedef float v8f __attribute__((ext_vector_type(8)));

constexpr int NB  = 8;     // batch
constexpr int NP  = 2048;  // points
constexpr int KNN = 20;    // neighbors
constexpr int NG  = 8;     // groupnorm groups
constexpr float EPS = 1e-5f;

// ---------------- WMMA helper: exact-f32 16x16x4 MAC -----------------------
// A(16x4): element(m,k) -> lane = m, k = vgpr + 2*(lane/16)
// B(4x16): element(k,n) -> lane = n, k = vgpr + 2*(lane/16)
// D(16x16): element(i,j) -> i = vgpr + 8*(lane/16), j = lane%16
DEV v8f wmma4(v2f a, v2f b, v8f c) {
  return __builtin_amdgcn_wmma_f32_16x16x4_f32(false, a, false, b, (short)0, c, false, false);
}

DEV float lrelu_f(float x) { return x >= 0.f ? x : 0.2f * x; }

// ---------------- utility kernels ------------------------------------------
__global__ __launch_bounds__(256) void zero_kernel(float* p, int n) {
  int t = blockIdx.x * blockDim.x + threadIdx.x;
  if (t < n) p[t] = 0.f;
}

// x (B,N,3) -> xT (B,N,4) with zero-padded 4th channel (point-major layout)
__global__ __launch_bounds__(256) void tr_kernel(const float* __restrict__ x,
                                                 float* __restrict__ xT) {
  int tid = blockIdx.x * 256 + threadIdx.x;
  if (tid >= NB * NP) return;
  const float* src = x + (size_t)tid * 3;
  v4f val = {src[0], src[1], src[2], 0.f};
  *(v4f*)(xT + (size_t)tid * 4) = val;
}

// w0 (64,6) -> wpad (64,8): cols {0,1,2,0, w3,w4,w5,0}
__global__ __launch_bounds__(64) void padw_kernel(const float* __restrict__ w,
                                                  float* __restrict__ wpad) {
  int o = threadIdx.x;
  const float* s = w + (size_t)o * 6;
  float* d = wpad + (size_t)o * 8;
  d[0] = s[0]; d[1] = s[1]; d[2] = s[2]; d[3] = 0.f;
  d[4] = s[3]; d[5] = s[4]; d[6] = s[5]; d[7] = 0.f;
}

// per-point sum of squares over channels: xx[b][n]  (X is point-major, LDC row stride)
template <int C, int LDC>
__global__ __launch_bounds__(256) void rss_kernel(const float* __restrict__ X,
                                                  float* __restrict__ xx) {
  int tid = blockIdx.x * 256 + threadIdx.x;
  if (tid >= NB * NP) return;
  const float* row = X + (size_t)tid * LDC;
  float s = 0.f;
#pragma unroll 8
  for (int c = 0; c < C; ++c) { float v = row[c]; s += v * v; }
  xx[tid] = s;
}

// ---------------- Gram / negative squared distances via f32 WMMA -----------
// one wave -> one 16x16 tile of pd ; pd = 2*X X^T - xx_n - xx_m
template <int C, int LDC>
__global__ __launch_bounds__(256) void gram_kernel(const float* __restrict__ X, int b,
                                                   const float* __restrict__ xx,
                                                   float* __restrict__ pd) {
  int wid  = (blockIdx.x * 256 + threadIdx.x) >> 5;  // 0..16383 (128x128 tiles)
  int lane = threadIdx.x & 31;
  int tn = wid >> 7, tm = wid & 127;
  int n0 = tn * 16, m0 = tm * 16;
  int l = lane & 15, half = lane >> 4;
  const float* Xb = X + (size_t)b * NP * LDC;
  const float* rowN = Xb + (size_t)(n0 + l) * LDC + 2 * half;
  const float* rowM = Xb + (size_t)(m0 + l) * LDC + 2 * half;
  v8f acc;
#pragma unroll
  for (int r = 0; r < 8; ++r) acc[r] = 0.f;
#pragma unroll
  for (int cb = 0; cb < C; cb += 4) {
    v2f av = *(const v2f*)(rowN + cb);
    v2f bv = *(const v2f*)(rowM + cb);
    acc = wmma4(av, bv, acc);
  }
  const float* xxb = xx + (size_t)b * NP;
#pragma unroll
  for (int r = 0; r < 8; ++r) {
    int n = n0 + r + 8 * half, m = m0 + l;
    pd[(size_t)n * NP + m] = 2.f * acc[r] - xxb[n] - xxb[m];
  }
}

// ---------------- top-k (k=20) per row: 20 argmax sweeps over LDS ----------
__global__ __launch_bounds__(256) void topk_kernel(const float* __restrict__ pd,
                                                   int* __restrict__ idx, int b) {
  __shared__ float vals[NP];
  __shared__ float sv[256];
  __shared__ int   si[256];
  int n = blockIdx.x, t = threadIdx.x;
  const float* row = pd + (size_t)n * NP;
  for (int i = t; i < NP; i += 256) vals[i] = row[i];
  __syncthreads();
  int* orow = idx + ((size_t)b * NP + n) * KNN;
  for (int s = 0; s < KNN; ++s) {
    float bvv = -3.4e38f; int bi = NP;
    for (int i = t; i < NP; i += 256) {
      float v = vals[i];
      if (v > bvv || (v == bvv && i < bi)) { bvv = v; bi = i; }
    }
    sv[t] = bvv; si[t] = bi;
    __syncthreads();
    for (int off = 128; off > 0; off >>= 1) {
      if (t < off) {
        float v2 = sv[t + off]; int i2 = si[t + off];
        if (v2 > sv[t] || (v2 == sv[t] && i2 < si[t])) { sv[t] = v2; si[t] = i2; }
      }
      __syncthreads();
    }
    if (t == 0) { orow[s] = si[0]; vals[si[0]] = -3.4e38f; }
    __syncthreads();
  }
}

// ---------------- edge conv: gathered GEMM via WMMA ------------------------
// one wave -> 16(o) x 80(nk) strip (= 4 points x 20 neighbors, 5 WMMA tiles).
// Fused K-loop: each gathered (fm, fn) pair feeds TWO WMMAs (edge cols + ctr cols).
// emits: ymax/ymin over k per (b,n,o), and per-(b,group) sum/sumsq via atomics.
template <int C, int O, int GS, int LDC>
__global__ __launch_bounds__(256) void edgeconv_kernel(
    const float* __restrict__ X, const int* __restrict__ idx,
    const float* __restrict__ w, const float* __restrict__ bias,
    float* __restrict__ ymax, float* __restrict__ ymin,
    float* __restrict__ gsum, float* __restrict__ gsumsq) {
  constexpr int C2 = 2 * C;
  __shared__ float tile[8][16][80];
  int wslot = threadIdx.x >> 5;
  int lane  = threadIdx.x & 31;
  int wid = blockIdx.x * 8 + wslot;
  constexpr int wavesPerB = (O / 16) * (NP * KNN / 80);   // (O/16)*512
  int b  = wid / wavesPerB;
  int r0 = wid % wavesPerB;
  int ot = r0 >> 9, strip = r0 & 511;
  int o0 = ot * 16, nk0 = strip * 80;
  int l = lane & 15, half = lane >> 4;
  const float* Xb  = X + (size_t)b * NP * LDC;
  const int* idxb  = idx + (size_t)b * NP * KNN;

  const float* rowN[5];
  const float* rowM[5];
#pragma unroll
  for (int t = 0; t < 5; ++t) {
    int nk = nk0 + 16 * t + l;
    int n = nk / 20, kk = nk - n * 20;
    int m = idxb[n * KNN + kk];
    rowN[t] = Xb + (size_t)n * LDC + 2 * half;
    rowM[t] = Xb + (size_t)m * LDC + 2 * half;
  }
  const float* wrow = w + (size_t)(o0 + l) * C2 + 2 * half;

  float bv[8];
#pragma unroll
  for (int r = 0; r < 8; ++r) bv[r] = bias[o0 + r + 8 * half];
  v8f acc[5];
#pragma unroll
  for (int t = 0; t < 5; ++t)
#pragma unroll
    for (int r = 0; r < 8; ++r) acc[t][r] = bv[r];

#pragma unroll 4
  for (int cb = 0; cb < C; cb += 4) {
    v2f awe = *(const v2f*)(wrow + cb);        // weights for (nbr - ctr) columns
    v2f awc = *(const v2f*)(wrow + C + cb);    // weights for ctr columns
#pragma unroll
    for (int t = 0; t < 5; ++t) {
      v2f fm = *(const v2f*)(rowM[t] + cb);
      v2f fn = *(const v2f*)(rowN[t] + cb);
      acc[t] = wmma4(awe, fm - fn, acc[t]);
      acc[t] = wmma4(awc, fn, acc[t]);
    }
  }

  float s = 0.f, q = 0.f;
#pragma unroll
  for (int t = 0; t < 5; ++t)
#pragma unroll
    for (int r = 0; r < 8; ++r) {
      float yv = acc[t][r];
      s += yv; q += yv * yv;
      tile[wslot][r + 8 * half][16 * t + l] = yv;
    }
  __syncthreads();

  // max/min over k: 16 o x 4 n pairs, 2 per lane; ymax/ymin layout (b, n, o)
  for (int p = lane; p < 64; p += 32) {
    int o_loc = p >> 2, n_loc = p & 3;
    float mx = -3.4e38f, mn = 3.4e38f;
#pragma unroll
    for (int kk = 0; kk < KNN; ++kk) {
      float v = tile[wslot][o_loc][n_loc * KNN + kk];
      mx = fmaxf(mx, v); mn = fminf(mn, v);
    }
    int n = strip * 4 + n_loc, o = o0 + o_loc;
    ymax[((size_t)b * NP + n) * O + o] = mx;
    ymin[((size_t)b * NP + n) * O + o] = mn;
  }

  // group stats: butterfly within 16-lane halves (group boundary = lane half for GS=8)
#pragma unroll
  for (int d = 1; d < 16; d <<= 1) { s += __shfl_xor(s, d); q += __shfl_xor(q, d); }
  if constexpr (GS >= 16) {
    s += __shfl_xor(s, 16); q += __shfl_xor(q, 16);
    if (lane == 0) {
      atomicAdd(&gsum[b * NG + o0 / GS], s);
      atomicAdd(&gsumsq[b * NG + o0 / GS], q);
    }
  } else {  // GS == 8
    if (lane == 0)  { atomicAdd(&gsum[b * NG + o0 / 8], s);     atomicAdd(&gsumsq[b * NG + o0 / 8], q); }
    if (lane == 16) { atomicAdd(&gsum[b * NG + o0 / 8 + 1], s); atomicAdd(&gsumsq[b * NG + o0 / 8 + 1], q); }
  }
}

// ---------------- GN finalize for edge blocks (monotonicity trick) ---------
// tid = ((b*NP + n)*O + o): o fastest -> coalesced writes into xc (point-major)
__global__ __launch_bounds__(256) void gnfin_kernel(
    const float* __restrict__ ymax, const float* __restrict__ ymin,
    const float* __restrict__ gsum, const float* __restrict__ gsumsq,
    const float* __restrict__ gamma, const float* __restrict__ beta,
    float* __restrict__ xc, int O, int GS, int coff) {
  int tid = blockIdx.x * 256 + threadIdx.x;
  if (tid >= NB * O * NP) return;
  int o = tid % O;
  int n = (tid / O) % NP;
  int b = tid / (O * NP);
  int g = o / GS;
  float cnt = (float)GS * (float)NP * (float)KNN;
  float mu  = gsum[b * NG + g] / cnt;
  float var = gsumsq[b * NG + g] / cnt - mu * mu;
  float rs  = rsqrtf(var + EPS);
  float slope = rs * gamma[o];
  float v = (slope >= 0.f) ? ymax[tid] : ymin[tid];
  float yn = (v - mu) * slope + beta[o];
  xc[((size_t)b * NP + n) * 256 + coff + o] = lrelu_f(yn);
}

// ---------------- pointwise conv 256x256 via WMMA + group stats ------------
__global__ __launch_bounds__(256) void pconv_kernel(
    const float* __restrict__ xc, const float* __restrict__ w,
    const float* __restrict__ bias, float* __restrict__ y3,
    float* __restrict__ gsum, float* __restrict__ gsumsq) {
  int wid  = (blockIdx.x * 256 + threadIdx.x) >> 5;  // 8b x 16ot x 128mt
  int lane = threadIdx.x & 31;
  int b  = wid / (16 * 128);
  int r0 = wid % (16 * 128);
  int ot = r0 >> 7, mt = r0 & 127;
  int o0 = ot * 16, m0 = mt * 16;
  int l = lane & 15, half = lane >> 4;
  const float* Xrow = xc + ((size_t)b * NP + m0 + l) * 256 + 2 * half;
  const float* wrow = w + (size_t)(o0 + l) * 256 + 2 * half;
  v8f acc;
#pragma unroll
  for (int r = 0; r < 8; ++r) acc[r] = bias[o0 + r + 8 * half];
#pragma unroll 8
  for (int cb = 0; cb < 256; cb += 4) {
    v2f av = *(const v2f*)(wrow + cb);
    v2f fv = *(const v2f*)(Xrow + cb);
    acc = wmma4(av, fv, acc);
  }
  float s = 0.f, q = 0.f;
#pragma unroll
  for (int r = 0; r < 8; ++r) {
    float yv = acc[r];
    y3[((size_t)b * 256 + o0 + r + 8 * half) * NP + m0 + l] = yv;
    s += yv; q += yv * yv;
  }
#pragma unroll
  for (int d = 1; d < 32; d <<= 1) { s += __shfl_xor(s, d); q += __shfl_xor(q, d); }
  if (lane == 0) {  // GS=32 -> whole 16-o tile in one group
    atomicAdd(&gsum[b * NG + o0 / 32], s);
    atomicAdd(&gsumsq[b * NG + o0 / 32], q);
  }
}

// ---------------- GN + lrelu + max/mean pool over N ------------------------
__global__ __launch_bounds__(256) void gnpool_kernel(
    const float* __restrict__ y3,
    const float* __restrict__ gsum, const float* __restrict__ gsumsq,
    const float* __restrict__ gamma, const float* __restrict__ beta,
    float* __restrict__ vfeat) {
  __shared__ float smx[256], ssm[256];
  int b = blockIdx.x >> 8;
  int o = blockIdx.x & 255;
  int t = threadIdx.x;
  int g = o >> 5;
  float cnt = 32.f * (float)NP;
  float mu  = gsum[b * NG + g] / cnt;
  float var = gsumsq[b * NG + g] / cnt - mu * mu;
  float rs  = rsqrtf(var + EPS);
  float ga = gamma[o], be = beta[o];
  const float* row = y3 + ((size_t)b * 256 + o) * NP;
  float mx = -3.4e38f, sm = 0.f;
  for (int i = t; i < NP; i += 256) {
    float yn = (row[i] - mu) * rs * ga + be;
    yn = lrelu_f(yn);
    mx = fmaxf(mx, yn);
    sm += yn;
  }
  smx[t] = mx; ssm[t] = sm;
  __syncthreads();
  for (int off = 128; off > 0; off >>= 1) {
    if (t < off) { smx[t] = fmaxf(smx[t], smx[t + off]); ssm[t] += ssm[t + off]; }
    __syncthreads();
  }
  if (t == 0) {
    vfeat[b * 512 + o]       = smx[0];
    vfeat[b * 512 + 256 + o] = ssm[0] / (float)NP;
  }
}

// ---------------- tiny FC layers (M=8) -------------------------------------
__global__ __launch_bounds__(256) void fc_kernel(
    const float* __restrict__ in, const float* __restrict__ W,
    const float* __restrict__ bias, float* __restrict__ outp,
    int M, int Nout, int Kd, int doLrelu) {
  int tid = blockIdx.x * 256 + threadIdx.x;
  if (tid >= M * Nout) return;
  int m = tid / Nout, o = tid % Nout;
  float s = bias[o];
  const float* iv = in + (size_t)m * Kd;
  const float* wv = W + (size_t)o * Kd;
#pragma unroll 4
  for (int k = 0; k < Kd; ++k) s += iv[k] * wv[k];
  if (doLrelu && s < 0.f) s *= 0.2f;
  outp[tid] = s;
}

// ---------------- host orchestration ---------------------------------------
extern "C" void kernel_launch(void* const* d_in, const int* in_sizes, int n_in,
                              void* d_out, int out_size, void* d_ws, size_t ws_size,
                              hipStream_t stream) {
  (void)in_sizes; (void)n_in; (void)out_size; (void)ws_size;
  const float* x   = (const float*)d_in[0];
  const float* w0  = (const float*)d_in[1];
  const float* b0  = (const float*)d_in[2];
  const float* g0  = (const float*)d_in[3];
  const float* h0  = (const float*)d_in[4];
  const float* w1  = (const float*)d_in[5];
  const float* b1  = (const float*)d_in[6];
  const float* g1  = (const float*)d_in[7];
  const float* h1  = (const float*)d_in[8];
  const float* w2  = (const float*)d_in[9];
  const float* b2  = (const float*)d_in[10];
  const float* g2  = (const float*)d_in[11];
  const float* h2  = (const float*)d_in[12];
  const float* w3  = (const float*)d_in[13];
  const float* b3  = (const float*)d_in[14];
  const float* g3  = (const float*)d_in[15];
  const float* h3  = (const float*)d_in[16];
  const float* fw0 = (const float*)d_in[17];
  const float* fb0 = (const float*)d_in[18];
  const float* fw1 = (const float*)d_in[19];
  const float* fb1 = (const float*)d_in[20];
  const float* fw2 = (const float*)d_in[21];
  const float* fb2 = (const float*)d_in[22];
  float* out = (float*)d_out;

  // workspace layout (floats); pd and y3 alias (both 4M floats, disjoint in time)
  float* ws = (float*)d_ws;
  size_t off = 0;
  float* xT    = ws + off; off += (size_t)NB * NP * 4;    // point-major, padded C=4
  float* wpad  = ws + off; off += 64 * 8;
  float* xx    = ws + off; off += (size_t)NB * NP;
  float* pd    = ws + off; off += (size_t)NP * NP;        // per-batch distances / y3
  float* y3    = pd;
  float* xc    = ws + off; off += (size_t)NB * NP * 256;  // point-major features
  float* ymax  = ws + off; off += (size_t)NB * NP * 128;  // (b, n, o)
  float* ymin  = ws + off; off += (size_t)NB * NP * 128;
  float* gsum  = ws + off; off += 64;
  float* gsq   = ws + off; off += 64;
  float* vfeat = ws + off; off += (size_t)NB * 512;
  float* t0    = ws + off; off += (size_t)NB * 512;
  float* t1    = ws + off; off += (size_t)NB * 512;
  int*   idxp  = (int*)(ws + off);

  tr_kernel<<<(NB * NP + 255) / 256, 256, 0, stream>>>(x, xT);
  padw_kernel<<<1, 64, 0, stream>>>(w0, wpad);

  // ---- stage 0: C=4(padded), O=64, GS=8, input xT (LDC=4) ----
  rss_kernel<4, 4><<<(NB * NP + 255) / 256, 256, 0, stream>>>(xT, xx);
  for (int b = 0; b < NB; ++b) {
    gram_kernel<4, 4><<<2048, 256, 0, stream>>>(xT, b, xx, pd);
    topk_kernel<<<NP, 256, 0, stream>>>(pd, idxp, b);
  }
  zero_kernel<<<1, 128, 0, stream>>>(gsum, 128);
  edgeconv_kernel<4, 64, 8, 4><<<NB * 4 * 512 / 8, 256, 0, stream>>>(
      xT, idxp, wpad, b0, ymax, ymin, gsum, gsq);
  gnfin_kernel<<<(NB * 64 * NP + 255) / 256, 256, 0, stream>>>(
      ymax, ymin, gsum, gsq, g0, h0, xc, 64, 8, 0);

  // ---- stage 1: C=64, O=64, GS=8, input xc cols [0,64) (LDC=256) ----
  rss_kernel<64, 256><<<(NB * NP + 255) / 256, 256, 0, stream>>>(xc, xx);
  for (int b = 0; b < NB; ++b) {
    gram_kernel<64, 256><<<2048, 256, 0, stream>>>(xc, b, xx, pd);
    topk_kernel<<<NP, 256, 0, stream>>>(pd, idxp, b);
  }
  zero_kernel<<<1, 128, 0, stream>>>(gsum, 128);
  edgeconv_kernel<64, 64, 8, 256><<<NB * 4 * 512 / 8, 256, 0, stream>>>(
      xc, idxp, w1, b1, ymax, ymin, gsum, gsq);
  gnfin_kernel<<<(NB * 64 * NP + 255) / 256, 256, 0, stream>>>(
      ymax, ymin, gsum, gsq, g1, h1, xc, 64, 8, 64);

  // ---- stage 2: C=64, O=128, GS=16, input xc cols [64,128) (LDC=256) ----
  rss_kernel<64, 256><<<(NB * NP + 255) / 256, 256, 0, stream>>>(xc + 64, xx);
  for (int b = 0; b < NB; ++b) {
    gram_kernel<64, 256><<<2048, 256, 0, stream>>>(xc + 64, b, xx, pd);
    topk_kernel<<<NP, 256, 0, stream>>>(pd, idxp, b);
  }
  zero_kernel<<<1, 128, 0, stream>>>(gsum, 128);
  edgeconv_kernel<64, 128, 16, 256><<<NB * 8 * 512 / 8, 256, 0, stream>>>(
      xc + 64, idxp, w2, b2, ymax, ymin, gsum, gsq);
  gnfin_kernel<<<(NB * 128 * NP + 255) / 256, 256, 0, stream>>>(
      ymax, ymin, gsum, gsq, g2, h2, xc, 128, 16, 128);

  // ---- stage 3: pointwise conv + GN + pool + MLP ----
  zero_kernel<<<1, 128, 0, stream>>>(gsum, 128);
  pconv_kernel<<<2048, 256, 0, stream>>>(xc, w3, b3, y3, gsum, gsq);
  gnpool_kernel<<<NB * 256, 256, 0, stream>>>(y3, gsum, gsq, g3, h3, vfeat);

  fc_kernel<<<(NB * 512 + 255) / 256, 256, 0, stream>>>(vfeat, fw0, fb0, t0, NB, 512, 512, 1);
  fc_kernel<<<(NB * 512 + 255) / 256, 256, 0, stream>>>(t0,    fw1, fb1, t1, NB, 512, 512, 1);
  fc_kernel<<<(NB * 256 + 255) / 256, 256, 0, stream>>>(t1,    fw2, fb2, out, NB, 256, 512, 0);
}